// GraphResNetBlock_43396349559144
// MI455X (gfx1250) — compile-verified
//
#include <hip/hip_runtime.h>
#include <stdint.h>

typedef __attribute__((ext_vector_type(2))) float v2f;
typedef __attribute__((ext_vector_type(8))) float v8f;

#define N_NODES 50000
#define N_EDGES 800000
#define DFEAT   128

// ---------------- degree / normalization ----------------

__global__ void k_deg_init(uint32_t* __restrict__ deg) {
    int i = blockIdx.x * blockDim.x + threadIdx.x;
    if (i < N_NODES) deg[i] = 1u;   // self loop contributes 1
}

__global__ void k_deg_count(const long long* __restrict__ col, uint32_t* __restrict__ deg) {
    int e = blockIdx.x * blockDim.x + threadIdx.x;
    if (e < N_EDGES) atomicAdd(&deg[(int)col[e]], 1u);
}

__global__ void k_dis(const uint32_t* __restrict__ deg, float* __restrict__ dis) {
    int i = blockIdx.x * blockDim.x + threadIdx.x;
    if (i < N_NODES) dis[i] = rsqrtf((float)deg[i]);   // deg >= 1 always
}

// ---------------- h = x @ W via f32 WMMA (16x16x4) ----------------
// One wave per 16x16 output tile. 50000/16 = 3125 row tiles, 128/16 = 8 col
// tiles -> 25000 waves, 8 waves/block -> 3125 blocks (exact, no partial waves,
// EXEC stays all-ones for WMMA).
__global__ void k_gemm(const float* __restrict__ x, const float* __restrict__ W,
                       float* __restrict__ h) {
    const int wave = (blockIdx.x * blockDim.x + threadIdx.x) >> 5;
    const int lane = threadIdx.x & 31;
    const int tilesN = DFEAT / 16;           // 8
    const int tm = wave / tilesN;
    const int tn = wave % tilesN;
    const int half = lane >> 4;              // 0: lanes 0-15, 1: lanes 16-31
    const int lrow = lane & 15;

    const int arow = tm * 16 + lrow;         // A-matrix row (M) for this lane
    const int bcol = tn * 16 + lrow;         // B-matrix col (N) for this lane

    v8f c = {};
    #pragma unroll
    for (int k0 = 0; k0 < DFEAT; k0 += 4) {
        // A 16x4 f32: VGPR0 = K = k0+2*half, VGPR1 = K+1 (ISA 32-bit A layout)
        const int ka = k0 + 2 * half;
        v2f a, b;
        a.x = x[arow * DFEAT + ka];
        a.y = x[arow * DFEAT + ka + 1];
        // B 4x16 f32: row K striped across lanes within a VGPR
        b.x = W[(ka + 0) * DFEAT + bcol];
        b.y = W[(ka + 1) * DFEAT + bcol];
        c = __builtin_amdgcn_wmma_f32_16x16x4_f32(
                /*neg_a=*/false, a, /*neg_b=*/false, b,
                /*c_mod=*/(short)0, c, /*reuse_a=*/false, /*reuse_b=*/false);
    }
    // C/D layout: VGPR i -> M = i + 8*half, N = lrow
    #pragma unroll
    for (int i = 0; i < 8; ++i) {
        h[(tm * 16 + i + 8 * half) * DFEAT + bcol] = c[i];
    }
}

// ---------------- self-loop term: acc = dis[i]^2 * h[i] ----------------
__global__ void k_acc_init(const float* __restrict__ h, const float* __restrict__ dis,
                           float* __restrict__ acc) {
    int idx = blockIdx.x * blockDim.x + threadIdx.x;     // over N*D/4 float4s
    if (idx < N_NODES * DFEAT / 4) {
        int node = idx / (DFEAT / 4);
        float s = dis[node];
        s = s * s;                                        // norm of self loop
        float4 hv = ((const float4*)h)[idx];
        float4 o = make_float4(hv.x * s, hv.y * s, hv.z * s, hv.w * s);
        ((float4*)acc)[idx] = o;
    }
}

// ---------------- edge scatter: acc[col] += dis[row]*dis[col] * h[row] ----------------
// One wave per edge; each lane owns 4 consecutive features (32*4 = 128).
__global__ void k_scatter(const long long* __restrict__ row,
                          const long long* __restrict__ col,
                          const float* __restrict__ h, const float* __restrict__ dis,
                          float* __restrict__ acc) {
    const int e = (blockIdx.x * blockDim.x + threadIdx.x) >> 5;
    const int lane = threadIdx.x & 31;
    if (e >= N_EDGES) return;                 // wave-uniform; grid is exact anyway
    const int r = (int)row[e];
    const int c = (int)col[e];
    const float norm = dis[r] * dis[c];
    const float4 hv = ((const float4*)(h + (size_t)r * DFEAT))[lane];
    float* dst = acc + (size_t)c * DFEAT + lane * 4;
    atomicAdd(dst + 0, norm * hv.x);
    atomicAdd(dst + 1, norm * hv.y);
    atomicAdd(dst + 2, norm * hv.z);
    atomicAdd(dst + 3, norm * hv.w);
}

// ---------------- out = tanh(acc) + x ----------------
__global__ void k_final(const float* __restrict__ x, float* __restrict__ out) {
    int idx = blockIdx.x * blockDim.x + threadIdx.x;     // over N*D/4 float4s
    if (idx < N_NODES * DFEAT / 4) {
        float4 a = ((const float4*)out)[idx];
        float4 xv = ((const float4*)x)[idx];
        float4 o = make_float4(tanhf(a.x) + xv.x, tanhf(a.y) + xv.y,
                               tanhf(a.z) + xv.z, tanhf(a.w) + xv.w);
        ((float4*)out)[idx] = o;
    }
}

extern "C" void kernel_launch(void* const* d_in, const int* in_sizes, int n_in,
                              void* d_out, int out_size, void* d_ws, size_t ws_size,
                              hipStream_t stream) {
    const float*     x  = (const float*)d_in[0];                 // [N, D]
    const long long* ei = (const long long*)d_in[1];             // [2, E] int64
    const float*     W  = (const float*)d_in[2];                 // [D, D]
    const long long* row = ei;                                   // source nodes
    const long long* col = ei + N_EDGES;                         // target nodes
    float* out = (float*)d_out;                                  // acc, then final

    // workspace layout: h [N*D f32] | dis [N f32] | deg [N u32]
    float*    h   = (float*)d_ws;
    float*    dis = h + (size_t)N_NODES * DFEAT;
    uint32_t* deg = (uint32_t*)(dis + N_NODES);

    const int TPB = 256;

    k_deg_init <<<(N_NODES + TPB - 1) / TPB, TPB, 0, stream>>>(deg);
    k_deg_count<<<(N_EDGES + TPB - 1) / TPB, TPB, 0, stream>>>(col, deg);
    k_dis      <<<(N_NODES + TPB - 1) / TPB, TPB, 0, stream>>>(deg, dis);

    // 25000 tile-waves, 8 waves per 256-thread block -> 3125 blocks exact
    k_gemm<<<(N_NODES / 16) * (DFEAT / 16) / (TPB / 32), TPB, 0, stream>>>(x, W, h);

    k_acc_init<<<(N_NODES * DFEAT / 4 + TPB - 1) / TPB, TPB, 0, stream>>>(h, dis, out);

    // one wave per edge: 800000 waves / 8 per block -> 100000 blocks exact
    k_scatter<<<N_EDGES / (TPB / 32), TPB, 0, stream>>>(row, col, h, dis, out);

    k_final<<<(N_NODES * DFEAT / 4 + TPB - 1) / TPB, TPB, 0, stream>>>(x, out);
}